// LightGATLayer_49933289783544
// MI455X (gfx1250) — compile-verified
//
#include <hip/hip_runtime.h>
#include <math.h>

#define N_NODES   100000
#define N_EDGES   1000000
#define EN        (N_EDGES + N_NODES)
#define D         64
#define NEG_SLOPE 0.2f
#define LN_EPS    1e-5f

typedef __attribute__((ext_vector_type(2))) float v2f;
typedef __attribute__((ext_vector_type(8))) float v8f;

// ---------------- float <-> order-preserving uint (for atomic segment-max) ---
__device__ __forceinline__ unsigned f2ord(float f) {
    unsigned u = __float_as_uint(f);
    return (u & 0x80000000u) ? ~u : (u | 0x80000000u);
}
__device__ __forceinline__ float ord2f(unsigned u) {
    u = (u & 0x80000000u) ? (u & 0x7FFFFFFFu) : ~u;
    return __uint_as_float(u);
}

// ---------------- Kernel 1: x_l = x@W_l + b_l ; x_r = x@W_r + b_r (WMMA fp32)
// One wave (32 threads) per 16-row tile; gridDim.y selects W_l / W_r.
// Uses V_WMMA_F32_16X16X4_F32: A = 16x4 fp32 (2 VGPRs/lane), B = 4x16 fp32,
// C/D = 16x16 fp32 in 8 VGPRs.  K=64 -> 16 k-steps x 4 col-tiles = 64 WMMAs.
__global__ __launch_bounds__(32)
void gat_gemm_wmma(const float* __restrict__ x,
                   const float* __restrict__ W_l, const float* __restrict__ b_l,
                   const float* __restrict__ W_r, const float* __restrict__ b_r,
                   float* __restrict__ x_l, float* __restrict__ x_r)
{
    const int rowBase = blockIdx.x << 4;
    const float* __restrict__ W  = blockIdx.y ? W_r : W_l;
    const float* __restrict__ bv = blockIdx.y ? b_r : b_l;
    float*       __restrict__ out = blockIdx.y ? x_r : x_l;

    const int lane = threadIdx.x;     // 0..31
    const int hi   = lane >> 4;       // lane half: K offset selector
    const int l15  = lane & 15;

    v8f c0 = {}, c1 = {}, c2 = {}, c3 = {};
    const float* arow = x + (size_t)(rowBase + l15) * D;

    for (int k = 0; k < D; k += 4) {
        // A 16x4: lanes 0-15 -> M=lane, K={k,k+1}; lanes 16-31 -> K={k+2,k+3}
        const int kk = k + (hi << 1);
        v2f a;
        a[0] = arow[kk];
        a[1] = arow[kk + 1];
        // B 4x16: VGPR0 -> rows K=kk, VGPR1 -> rows K=kk+1, cols = l15
        const float* w0 = W + (size_t)kk * D;
        const float* w1 = w0 + D;
        v2f b;
        b[0] = w0[l15];      b[1] = w1[l15];
        c0 = __builtin_amdgcn_wmma_f32_16x16x4_f32(false, a, false, b, (short)0, c0, false, false);
        b[0] = w0[16 + l15]; b[1] = w1[16 + l15];
        c1 = __builtin_amdgcn_wmma_f32_16x16x4_f32(false, a, false, b, (short)0, c1, false, false);
        b[0] = w0[32 + l15]; b[1] = w1[32 + l15];
        c2 = __builtin_amdgcn_wmma_f32_16x16x4_f32(false, a, false, b, (short)0, c2, false, false);
        b[0] = w0[48 + l15]; b[1] = w1[48 + l15];
        c3 = __builtin_amdgcn_wmma_f32_16x16x4_f32(false, a, false, b, (short)0, c3, false, false);
    }

    // C/D: VGPR r -> lanes 0-15: M=r, N=lane; lanes 16-31: M=r+8, N=lane-16
    const int mo = hi << 3;
#pragma unroll
    for (int r = 0; r < 8; ++r) {
        float* orow = out + (size_t)(rowBase + mo + r) * D;
        orow[     l15] = c0[r] + bv[     l15];
        orow[16 + l15] = c1[r] + bv[16 + l15];
        orow[32 + l15] = c2[r] + bv[32 + l15];
        orow[48 + l15] = c3[r] + bv[48 + l15];
    }
}

// ---------------- Kernel 2: per-edge attention logit + atomic segment-max ----
// Warp-per-edge: 32 lanes stream the two 256B rows coalesced (float2/lane).
__global__ __launch_bounds__(256)
void edge_logits(const int* __restrict__ ei, const float* __restrict__ x_l,
                 const float* __restrict__ x_r, const float* __restrict__ att,
                 float* __restrict__ e, unsigned* __restrict__ m_ord)
{
    const int warp = (blockIdx.x * blockDim.x + threadIdx.x) >> 5;
    const int lane = threadIdx.x & 31;
    if (warp >= EN) return;
    int src, dst;
    if (warp < N_EDGES) { src = ei[warp]; dst = ei[N_EDGES + warp]; }
    else                { src = dst = warp - N_EDGES; }   // self loop

    const float2 xl = *(const float2*)(x_l + (size_t)src * D + lane * 2);
    const float2 xr = *(const float2*)(x_r + (size_t)dst * D + lane * 2);
    const float2 at = *(const float2*)(att + lane * 2);
    float s0 = xl.x + xr.x; s0 = (s0 > 0.f) ? s0 : NEG_SLOPE * s0;
    float s1 = xl.y + xr.y; s1 = (s1 > 0.f) ? s1 : NEG_SLOPE * s1;
    float v = s0 * at.x + s1 * at.y;
#pragma unroll
    for (int m = 16; m >= 1; m >>= 1) v += __shfl_xor(v, m, 32);
    if (lane == 0) {
        e[warp] = v;
        atomicMax(&m_ord[dst], f2ord(v));
    }
}

// ---------------- Kernel 3: p = exp(e - m[dst]); z[dst] += p  ---------------
__global__ __launch_bounds__(256)
void edge_softmax_num(const int* __restrict__ ei, float* __restrict__ e,
                      const unsigned* __restrict__ m_ord, float* __restrict__ z)
{
    const int i = blockIdx.x * blockDim.x + threadIdx.x;
    if (i >= EN) return;
    const int dst = (i < N_EDGES) ? ei[N_EDGES + i] : (i - N_EDGES);
    const float p = expf(e[i] - ord2f(m_ord[dst]));
    e[i] = p;                       // overwrite logits with numerator in place
    atomicAdd(&z[dst], p);
}

// ---------------- Kernel 4: out[dst] += (p/z[dst]) * x_l[src]  --------------
__global__ __launch_bounds__(256)
void edge_aggregate(const int* __restrict__ ei, const float* __restrict__ p,
                    const float* __restrict__ z, const float* __restrict__ x_l,
                    float* __restrict__ out)
{
    const int warp = (blockIdx.x * blockDim.x + threadIdx.x) >> 5;
    const int lane = threadIdx.x & 31;
    if (warp >= EN) return;
    int src, dst;
    if (warp < N_EDGES) { src = ei[warp]; dst = ei[N_EDGES + warp]; }
    else                { src = dst = warp - N_EDGES; }

    const float alpha = p[warp] / z[dst];
    const float2 xl = *(const float2*)(x_l + (size_t)src * D + lane * 2);
    float* orow = out + (size_t)dst * D + lane * 2;
    atomicAdd(orow,     alpha * xl.x);
    atomicAdd(orow + 1, alpha * xl.y);
}

// ---------------- Kernel 5: out = LN(out + bias) * gamma + beta  ------------
// Warp-per-node; wave32 shuffle reduction for mean / E[x^2].
__global__ __launch_bounds__(256)
void bias_layernorm(float* __restrict__ out, const float* __restrict__ bias,
                    const float* __restrict__ gamma, const float* __restrict__ beta)
{
    const int node = (blockIdx.x * blockDim.x + threadIdx.x) >> 5;
    const int lane = threadIdx.x & 31;
    if (node >= N_NODES) return;

    float* row = out + (size_t)node * D + lane * 2;
    float2 v = *(float2*)row;
    const float2 bb = *(const float2*)(bias + lane * 2);
    v.x += bb.x; v.y += bb.y;

    float s  = v.x + v.y;
    float sq = v.x * v.x + v.y * v.y;
#pragma unroll
    for (int m = 16; m >= 1; m >>= 1) {
        s  += __shfl_xor(s,  m, 32);
        sq += __shfl_xor(sq, m, 32);
    }
    const float mean = s * (1.f / 64.f);
    const float var  = sq * (1.f / 64.f) - mean * mean;
    const float inv  = rsqrtf(var + LN_EPS);
    const float2 g  = *(const float2*)(gamma + lane * 2);
    const float2 be = *(const float2*)(beta  + lane * 2);
    float2 r;
    r.x = (v.x - mean) * inv * g.x + be.x;
    r.y = (v.y - mean) * inv * g.y + be.y;
    *(float2*)row = r;
}

// ---------------------------------------------------------------------------
extern "C" void kernel_launch(void* const* d_in, const int* in_sizes, int n_in,
                              void* d_out, int out_size, void* d_ws, size_t ws_size,
                              hipStream_t stream)
{
    const float* x     = (const float*)d_in[0];
    const int*   ei    = (const int*)  d_in[1];   // (2,E): [0..E) = src, [E..2E) = dst
    const float* W_l   = (const float*)d_in[2];
    const float* b_l   = (const float*)d_in[3];
    const float* W_r   = (const float*)d_in[4];
    const float* b_r   = (const float*)d_in[5];
    const float* att   = (const float*)d_in[6];
    const float* bias  = (const float*)d_in[7];
    const float* gamma = (const float*)d_in[8];
    const float* beta  = (const float*)d_in[9];

    // workspace layout
    float*    x_l   = (float*)d_ws;                         // N*64
    float*    x_r   = x_l + (size_t)N_NODES * D;            // N*64
    float*    e     = x_r + (size_t)N_NODES * D;            // EN (logits -> p)
    unsigned* m_ord = (unsigned*)(e + EN);                  // N
    float*    z     = (float*)(m_ord + N_NODES);            // N
    float*    out   = (float*)d_out;                        // N*64 (accumulator + result)

    hipMemsetAsync(out,   0, (size_t)N_NODES * D * sizeof(float), stream);
    hipMemsetAsync(m_ord, 0, (size_t)N_NODES * sizeof(unsigned),  stream);
    hipMemsetAsync(z,     0, (size_t)N_NODES * sizeof(float),     stream);

    dim3 ggrid(N_NODES / 16, 2);                 // 6250 row tiles x {W_l, W_r}
    gat_gemm_wmma<<<ggrid, 32, 0, stream>>>(x, W_l, b_l, W_r, b_r, x_l, x_r);

    const int warpsPerBlock = 8;                 // 256 threads = 8 wave32
    const int eblocks = (EN + warpsPerBlock - 1) / warpsPerBlock;
    edge_logits<<<eblocks, 256, 0, stream>>>(ei, x_l, x_r, att, e, m_ord);
    edge_softmax_num<<<(EN + 255) / 256, 256, 0, stream>>>(ei, e, m_ord, z);
    edge_aggregate<<<eblocks, 256, 0, stream>>>(ei, e, z, x_l, out);

    const int nblocks = (N_NODES + warpsPerBlock - 1) / warpsPerBlock;
    bias_layernorm<<<nblocks, 256, 0, stream>>>(out, bias, gamma, beta);
}